// GuidedImplicitPointSampler_25580825215054
// MI455X (gfx1250) — compile-verified
//
#include <hip/hip_runtime.h>
#include <math.h>

// MI455X / gfx1250 KNN(k=8, d=3) + radius mask.
//
// d2(q,t) = |q|^2 + |t|^2 - 2 q.t.  Per 16x16 tile via V_WMMA_F32_16X16X4_F32
// with the padded K=3 slot carrying |t|^2:
//   A (16x4, targets):  [tx, ty, tz, |t|^2]
//   B (4x16, queries):  [qx, qy, qz, -0.5]^T per column
//   D[m][n] = t_m . q_n - 0.5*|t_m|^2   =>  d2 = |q|^2 - 2*D
// Smallest-d2 == largest-D, so top-8 runs directly on WMMA output.
//
// C-layout (wave32): lane n(<16) & lane n+16 both hold column n (query n);
// low lanes get rows m=vgpr (targets +0..7), high lanes rows m=vgpr+8.
// Each lane keeps a lane-local descending top-8 of D over its half of the
// targets; lane pairs (n, n+16) merge at the end with one shuffle + bitonic
// merge network.
//
// Two-deep software pipeline (+unroll 2 for register rotation): the
// accumulator consumed by selection in iteration c was produced by the WMMA
// issued one iteration earlier, and every ds_load lands a full iteration
// before its consuming WMMA — LDS latency and XDL (WMMA) latency are both
// covered by independent selection VALU co-executing with the matrix pipe,
// and the unroll lets the compiler rotate acc_cur/acc_nxt without v_movs.

typedef __attribute__((ext_vector_type(2))) float v2f;
typedef __attribute__((ext_vector_type(8))) float v8f;

#define OCC_RADIUS 0.25f
#define TGT_STAGE  4096           // targets per LDS stage: 4096 * 16B = 64 KB
#define BLOCK      512            // 16 wave32 waves -> 256 queries per block
#define QPB        256            // queries per block

__global__ __launch_bounds__(BLOCK)
void knn8_wmma_kernel(const float* __restrict__ to_filter,   // [n_q][6]
                      const float* __restrict__ targets,     // [n_t][3]
                      float* __restrict__ out,               // [n_q][8]
                      int n_q, int n_t)
{
    __shared__ float4 lds_t[TGT_STAGE];    // (x, y, z, |t|^2)

    const int tid  = threadIdx.x;
    const int lane = tid & 31;
    const int wave = tid >> 5;
    const int half = lane >> 4;            // 0: lanes 0-15, 1: lanes 16-31
    const int l15  = lane & 15;

    // ---- Query setup: B matrix (4x16 KxN, f32), fixed for whole kernel ----
    // Layout: v0: lanes0-15 = K0 row, lanes16-31 = K2 row
    //         v1: lanes0-15 = K1 row, lanes16-31 = K3 row
    const int q  = blockIdx.x * QPB + wave * 16 + l15;
    const float qx = to_filter[q * 6 + 0];
    const float qy = to_filter[q * 6 + 1];
    const float qz = to_filter[q * 6 + 2];
    const float q2 = qx * qx + qy * qy + qz * qz;
    v2f B;
    B.x = half ? qz    : qx;
    B.y = half ? -0.5f : qy;

    // ---- A matrix per-lane LDS address (float2 granularity) ----
    // Target i occupies float2 slots {2i: (x,y), 2i+1: (z,|t|^2)}.
    // A 16x4 layout: low lanes need (x,y)=K0,K1; high lanes (z,|t|^2)=K2,K3.
    const float2* lds2 = (const float2*)lds_t;
    const int lds2_base = 2 * l15 + half;

    // ---- per-lane top-8 of D (descending; best[0] largest) ----
    float best[8];
#pragma unroll
    for (int i = 0; i < 8; ++i) best[i] = -__builtin_inff();

    const int n_stage = n_t / TGT_STAGE;
    for (int s = 0; s < n_stage; ++s) {
        __syncthreads();
        // Cooperative stage: load targets, precompute |t|^2 once per block.
#pragma unroll
        for (int i = tid; i < TGT_STAGE; i += BLOCK) {
            const int t = s * TGT_STAGE + i;
            const float tx = targets[t * 3 + 0];
            const float ty = targets[t * 3 + 1];
            const float tz = targets[t * 3 + 2];
            lds_t[i] = make_float4(tx, ty, tz, tx * tx + ty * ty + tz * tz);
        }
        __syncthreads();

        // Warm L2 for the next stage (global_prefetch_b8 on gfx1250).
        if (s + 1 < n_stage) {
            __builtin_prefetch(targets + (size_t)(s + 1) * TGT_STAGE * 3 + tid * 24, 0, 0);
        }

        constexpr int nchunk = TGT_STAGE / 16;   // 256
        constexpr int cmask  = nchunk - 1;

        // ---- two-deep pipeline prologue ----
        {
            const float2 a0 = lds2[lds2_base];           // chunk 0
            float2 a_nxt    = lds2[32 + lds2_base];      // chunk 1
            v2f A0; A0.x = a0.x; A0.y = a0.y;
            v8f z0 = {};
            v8f acc_cur = __builtin_amdgcn_wmma_f32_16x16x4_f32(
                false, A0, false, B, (short)0, z0, false, false);

#pragma unroll 2
            for (int c = 0; c < nchunk; ++c) {
                // WMMA for chunk c+1 (its ds_load completed an iteration ago;
                // wrapped index at the tail produces a dead result).
                v2f An; An.x = a_nxt.x; An.y = a_nxt.y;
                v8f zz = {};
                v8f acc_nxt = __builtin_amdgcn_wmma_f32_16x16x4_f32(
                    false, An, false, B, (short)0, zz, false, false);

                // ds_load for chunk c+2 (consumed by next iteration's WMMA).
                a_nxt = lds2[((c + 2) & cmask) * 32 + lds2_base];

                // ---- selection on chunk c (acc produced one iter ago) ----
                const float cmax =
                    fmaxf(fmaxf(fmaxf(acc_cur[0], acc_cur[1]),
                                fmaxf(acc_cur[2], acc_cur[3])),
                          fmaxf(fmaxf(acc_cur[4], acc_cur[5]),
                                fmaxf(acc_cur[6], acc_cur[7])));
                if (cmax > best[7]) {
#pragma unroll
                    for (int v = 0; v < 8; ++v) {
                        const float cand = acc_cur[v];
                        if (cand > best[7]) {
                            best[7] = cand;      // bubble to keep descending
#pragma unroll
                            for (int i = 6; i >= 0; --i) {
                                const float lo = fminf(best[i], best[i + 1]);
                                const float hi = fmaxf(best[i], best[i + 1]);
                                best[i] = hi; best[i + 1] = lo;
                            }
                        }
                    }
                }

                acc_cur = acc_nxt;
            }
        }
    }

    // ---- merge lane pair (n, n+16): two descending 8-lists -> top-8 ----
    float other[8];
#pragma unroll
    for (int i = 0; i < 8; ++i) other[i] = __shfl_xor(best[i], 16, 32);

    // best (desc) ++ reverse(other) is bitonic; split keeps the 8 largest,
    // then a 3-stage bitonic merge sorts them descending. Fully branchless.
    float h[8];
#pragma unroll
    for (int i = 0; i < 8; ++i) h[i] = fmaxf(best[i], other[7 - i]);
#define CE(i, j) { float lo = fminf(h[i], h[j]); float hi = fmaxf(h[i], h[j]); h[i] = hi; h[j] = lo; }
    CE(0, 4) CE(1, 5) CE(2, 6) CE(3, 7)
    CE(0, 2) CE(1, 3) CE(4, 6) CE(5, 7)
    CE(0, 1) CE(2, 3) CE(4, 5) CE(6, 7)
#undef CE

    // h descending in D == ascending in distance. Lanes 0-15 write the row.
    if (half == 0) {
        float d[8];
#pragma unroll
        for (int k = 0; k < 8; ++k)
            d[k] = sqrtf(fmaxf(q2 - 2.0f * h[k], 0.0f));
        const bool good = d[0] > OCC_RADIUS;
        float4 lo = good ? make_float4(d[0], d[1], d[2], d[3])
                         : make_float4(0.f, 0.f, 0.f, 0.f);
        float4 hi = good ? make_float4(d[4], d[5], d[6], d[7])
                         : make_float4(0.f, 0.f, 0.f, 0.f);
        float4* orow = (float4*)(out + (size_t)q * 8);
        orow[0] = lo;
        orow[1] = hi;
    }
}

extern "C" void kernel_launch(void* const* d_in, const int* in_sizes, int n_in,
                              void* d_out, int out_size, void* d_ws, size_t ws_size,
                              hipStream_t stream) {
    const float* to_filter = (const float*)d_in[0];   // [n_q][6] f32
    const float* targets   = (const float*)d_in[1];   // [n_t][3] f32
    float*       out       = (float*)d_out;           // [n_q][8] f32

    const int n_q = in_sizes[0] / 6;   // 32768
    const int n_t = in_sizes[1] / 3;   // 16384

    const int grid = n_q / QPB;        // 128 blocks of 512 threads (16 waves)
    knn8_wmma_kernel<<<grid, BLOCK, 0, stream>>>(to_filter, targets, out, n_q, n_t);
}